// TimeSlider_8847632630051
// MI455X (gfx1250) — compile-verified
//
#include <hip/hip_runtime.h>
#include <hip/hip_bf16.h>

// out[b, t] = trunc(x[b]) + t   for b in [0,B), t in [0,window)
// Store-bandwidth-bound streamer.
// Fast path (window==512): each 128-thread block writes 8 rows (16 KiB) with
// 8 global_store_b128 per lane sharing one address VGPR (immediate offsets).
// Per-row cost: 1 s_trunc_f32 + 4 v_add_f32 + 1 b128 store.

#define ROWS_PER_BLOCK 8
#define FAST_WINDOW 512

__global__ __launch_bounds__(128, 8) void TimeSlider_8847632630051_kernel(
    const float* __restrict__ x,
    const int* __restrict__ window_p,
    float* __restrict__ out,
    int B) {
  const int window = *window_p;                 // uniform -> s_load_b32
  const int row0 = blockIdx.x * ROWS_PER_BLOCK;

  if (window == FAST_WINDOW) {                  // uniform scalar branch
    const int t0 = (int)threadIdx.x << 2;       // 0,4,...,508
    const float f0 = (float)t0;
    const float r1 = f0 + 1.0f;
    const float r2 = f0 + 2.0f;
    const float r3 = f0 + 3.0f;
    // One address VGPR; per-row delta is a compile-time byte offset.
    float* __restrict__ p = out + (size_t)row0 * FAST_WINDOW + t0;

    if (row0 + ROWS_PER_BLOCK <= B) {
      // Hot path: straight-line, no per-row branches.
#pragma unroll
      for (int r = 0; r < ROWS_PER_BLOCK; ++r) {
        const float base = truncf(x[row0 + r]);  // merged s_load + s_trunc_f32
        float4 v;
        v.x = base + f0;
        v.y = base + r1;
        v.z = base + r2;
        v.w = base + r3;
        *reinterpret_cast<float4*>(p + r * FAST_WINDOW) = v;  // imm offset r*2048
      }
    } else {
      // Partial last tile (not taken for B=65536).
#pragma unroll
      for (int r = 0; r < ROWS_PER_BLOCK; ++r) {
        if (row0 + r < B) {
          const float base = truncf(x[row0 + r]);
          float4 v;
          v.x = base + f0;
          v.y = base + r1;
          v.z = base + r2;
          v.w = base + r3;
          *reinterpret_cast<float4*>(p + r * FAST_WINDOW) = v;
        }
      }
    }
    return;
  }

  // Generic fallback (window != 512): keep it tiny — never executed in the
  // reference configuration; suppress unroll/vectorize to avoid code bloat.
#pragma clang loop unroll(disable)
  for (int r = 0; r < ROWS_PER_BLOCK; ++r) {
    const int row = row0 + r;
    if (row >= B) break;
    const float base = truncf(x[row]);
    float* __restrict__ orow = out + (size_t)row * (size_t)window;
#pragma clang loop unroll(disable) vectorize(disable)
    for (int t = (int)threadIdx.x; t < window; t += (int)blockDim.x) {
      orow[t] = base + (float)t;
    }
  }
}

extern "C" void kernel_launch(void* const* d_in, const int* in_sizes, int n_in,
                              void* d_out, int out_size, void* d_ws, size_t ws_size,
                              hipStream_t stream) {
  const float* x = (const float*)d_in[0];        // [B, 1] float32
  const int* window_p = (const int*)d_in[1];     // scalar int (device, 1 elem)
  float* out = (float*)d_out;                    // [B, window, 1] float32

  const int B = in_sizes[0];                     // 65536 rows
  const unsigned grid = (unsigned)((B + ROWS_PER_BLOCK - 1) / ROWS_PER_BLOCK);
  dim3 block(128);                               // 4 wave32s
  TimeSlider_8847632630051_kernel<<<grid, block, 0, stream>>>(x, window_p, out, B);
}